// TopKTopPSampler_17824114278633
// MI455X (gfx1250) — compile-verified
//
#include <hip/hip_runtime.h>
#include <hip/hip_bf16.h>
#include <math.h>

// Top-k / top-p (nucleus) sampler, one workgroup per batch row.
// BATCH=256, VOCAB=128000, k in [1,64], p in [0.5,1.0].
// out[row] = (float)argmax over surviving tokens of (logit - log q).

#define BATCH  256
#define VOCAB  128000
#define V4     (VOCAB / 4)
#define NT     1024
#define NWAVES (NT / 32)
#define CAP    1024   // survivor gather capacity (generically <= ~64+bin)

typedef float f4 __attribute__((ext_vector_type(4)));

// Order-preserving float -> uint key (ascending key == ascending float)
__device__ __forceinline__ unsigned f2key(float f) {
  unsigned u = __float_as_uint(f);
  return (u & 0x80000000u) ? ~u : (u | 0x80000000u);
}
__device__ __forceinline__ float key2f(unsigned k) {
  unsigned u = (k & 0x80000000u) ? (k & 0x7fffffffu) : ~k;
  return __uint_as_float(u);
}

// Inclusive suffix-sum within a wave32 (barrier-free).
__device__ __forceinline__ unsigned waveInclSuffix(unsigned x, int lane) {
  #pragma unroll
  for (int off = 1; off < 32; off <<= 1) {
    unsigned v = __shfl_down(x, off, 32);
    if (lane + off < 32) x += v;
  }
  return x;
}

// Given a filled histogram of nb bins, find largest digit d with
// suffix_count(d) >= kneed; update *s_prefix/*s_kneed. ~3 barriers total.
__device__ __forceinline__ void select_digit(unsigned* hist, int nb, int shift,
                                             unsigned prefix, int kneed,
                                             unsigned* s_prefix, int* s_kneed,
                                             unsigned* waveTot, unsigned* waveLater) {
  const int tid  = threadIdx.x;
  const int lane = tid & 31;
  const int wv   = tid >> 5;
  const int bpt  = (nb >= NT) ? (nb / NT) : 1;  // bins per thread (4 or 1)
  const int b0   = tid * bpt;

  unsigned cs = 0u;
  if (b0 < nb)
    for (int b = b0; b < b0 + bpt; ++b) cs += hist[b];

  unsigned ws = waveInclSuffix(cs, lane);      // suffix within wave, incl own
  if (lane == 0) waveTot[wv] = ws;             // wave total
  __syncthreads();
  if (tid < 32) {
    unsigned wt  = waveTot[tid];
    unsigned suf = waveInclSuffix(wt, tid);    // suffix over waves, incl own
    waveLater[tid] = suf - wt;                 // strictly-later waves
  }
  __syncthreads();
  const unsigned later = waveLater[wv];
  const unsigned Sme   = ws + later;                        // suffix incl my chunk
  const unsigned wsn   = __shfl_down(ws, 1, 32);
  const unsigned Snext = (lane == 31) ? later : (wsn + later);

  if (b0 < nb && Sme >= (unsigned)kneed && Snext < (unsigned)kneed) {
    unsigned running = Snext;
    for (int b = b0 + bpt - 1; b >= b0; --b) {
      running += hist[b];
      if (running >= (unsigned)kneed) {
        *s_prefix = prefix | ((unsigned)b << shift);
        *s_kneed  = kneed - (int)(running - hist[b]);
        break;
      }
    }
  }
  __syncthreads();
}

__global__ __launch_bounds__(NT)
void topk_topp_sample_kernel(const float* __restrict__ logits,
                             const int*   __restrict__ kArr,
                             const float* __restrict__ pArr,
                             const float* __restrict__ qArr,
                             float*       __restrict__ out)
{
  const int row  = blockIdx.x;
  const int tid  = threadIdx.x;
  const int lane = tid & 31;
  const int wv   = tid >> 5;
  const f4* __restrict__ L4 = (const f4*)(logits + (size_t)row * VOCAB);
  const f4* __restrict__ Q4 = (const f4*)(qArr   + (size_t)row * VOCAB);

  __shared__ unsigned hist[4096];
  __shared__ float    cand[CAP];
  __shared__ float    srt[CAP];
  __shared__ unsigned waveTot[NWAVES];
  __shared__ unsigned waveLater[NWAVES];
  __shared__ float    rf[NWAVES];
  __shared__ int      ri[NWAVES];
  __shared__ unsigned s_prefix;
  __shared__ int      s_kneed;
  __shared__ int      s_cnt;
  __shared__ float    s_M, s_thr, s_tp;

  if (tid == 0) {
    s_prefix = 0u;
    int kk = kArr[row];
    s_kneed = (kk < 1) ? 1 : ((kk > VOCAB) ? VOCAB : kk);
    s_cnt = 0;
  }
  __syncthreads();

  float localMax = -INFINITY;

  // -------- Stage 0: histogram of key[31:20] (4096 bins); fused row-max ----
  for (int b = tid; b < 4096; b += NT) hist[b] = 0u;
  __syncthreads();
  {
    for (int i = tid; i < V4; i += NT) {
      f4 x = L4[i];
      localMax = fmaxf(localMax, fmaxf(fmaxf(x.x, x.y), fmaxf(x.z, x.w)));
      atomicAdd(&hist[f2key(x.x) >> 20], 1u);
      atomicAdd(&hist[f2key(x.y) >> 20], 1u);
      atomicAdd(&hist[f2key(x.z) >> 20], 1u);
      atomicAdd(&hist[f2key(x.w) >> 20], 1u);
    }
    __syncthreads();
    select_digit(hist, 4096, 20, 0u, s_kneed, &s_prefix, &s_kneed, waveTot, waveLater);
  }

  // -------- Stage 1: histogram of key[19:8] within prefix ------------------
  for (int b = tid; b < 4096; b += NT) hist[b] = 0u;
  __syncthreads();
  {
    const unsigned prefix = s_prefix;
    const int      kneed  = s_kneed;
    for (int i = tid; i < V4; i += NT) {
      f4 x = L4[i];
      #define PROC1(v) do {                                                  \
        unsigned key = f2key(v);                                             \
        if ((key & 0xFFF00000u) == prefix)                                   \
          atomicAdd(&hist[(key >> 8) & 0xFFFu], 1u);                         \
      } while (0)
      PROC1(x.x); PROC1(x.y); PROC1(x.z); PROC1(x.w);
      #undef PROC1
    }
    __syncthreads();
    select_digit(hist, 4096, 8, prefix, kneed, &s_prefix, &s_kneed, waveTot, waveLater);
  }

  // -------- Stage 2: histogram of key[7:0] within prefix; fused gather -----
  // Every final survivor has key >= prefix (top-24 bits known), so gather
  // that superset now and filter by the exact threshold afterwards.
  for (int b = tid; b < 256; b += NT) hist[b] = 0u;
  __syncthreads();
  {
    const unsigned prefix = s_prefix;   // bits [31:8] set, low byte 0
    const int      kneed  = s_kneed;
    for (int i = tid; i < V4; i += NT) {
      f4 x = L4[i];
      #define PROC2(v) do {                                                  \
        unsigned key = f2key(v);                                             \
        if (key >= prefix) {                                                 \
          int pos = atomicAdd(&s_cnt, 1);                                    \
          if (pos < CAP) cand[pos] = (v);                                    \
          if ((key & 0xFFFFFF00u) == prefix)                                 \
            atomicAdd(&hist[key & 0xFFu], 1u);                               \
        }                                                                    \
      } while (0)
      PROC2(x.x); PROC2(x.y); PROC2(x.z); PROC2(x.w);
      #undef PROC2
    }
    __syncthreads();
    select_digit(hist, 256, 0, prefix, kneed, &s_prefix, &s_kneed, waveTot, waveLater);
  }

  // -------- Row max + exact threshold --------------------------------------
  for (int off = 16; off > 0; off >>= 1)
    localMax = fmaxf(localMax, __shfl_xor(localMax, off, 32));
  if (lane == 0) rf[wv] = localMax;
  __syncthreads();
  if (tid == 0) {
    float m = rf[0];
    for (int w = 1; w < NWAVES; ++w) m = fmaxf(m, rf[w]);
    s_M   = m;
    s_thr = key2f(s_prefix);   // exact value of the k-th largest logit
  }
  __syncthreads();

  const float thr = s_thr;
  int cntC = s_cnt;

  // Pathological overflow (massive ties): exact re-gather pass. Uniform branch.
  if (cntC > CAP) {
    if (tid == 0) s_cnt = 0;
    __syncthreads();
    for (int i = tid; i < V4; i += NT) {
      f4 x = L4[i];
      #define PROCG(v) do {                                                  \
        if ((v) >= thr) {                                                    \
          int pos = atomicAdd(&s_cnt, 1);                                    \
          if (pos < CAP) cand[pos] = (v);                                    \
        }                                                                    \
      } while (0)
      PROCG(x.x); PROCG(x.y); PROCG(x.z); PROCG(x.w);
      #undef PROCG
    }
    __syncthreads();
    cntC = (s_cnt < CAP) ? s_cnt : CAP;
  }

  // -------- Rank-sort survivors (>= thr) ascending --------------------------
  for (int i = tid; i < cntC; i += NT) {
    float v = cand[i];
    if (v >= thr) {
      int rank = 0;
      for (int j = 0; j < cntC; ++j) {
        float vj = cand[j];
        rank += (vj >= thr) && ((vj < v) || (vj == v && j < i));
      }
      srt[rank] = v;
    }
  }
  __syncthreads();

  // -------- Nucleus cut: drop ascending prefix with cum mass <= (1-p) ------
  if (tid == 0) {
    int ns = 0;
    for (int j = 0; j < cntC; ++j) ns += (cand[j] >= thr);
    const float M = s_M;
    const float p = pArr[row];
    float S = 0.f;
    for (int i = 0; i < ns; ++i) S += expf(srt[i] - M);
    const float target = (1.0f - p) * S;
    float c = 0.f;
    float tp = srt[ns - 1];
    for (int i = 0; i < ns; ++i) {
      c += expf(srt[i] - M);
      if (c > target) { tp = srt[i]; break; }
    }
    s_tp = tp;
  }
  __syncthreads();

  // -------- Final pass: argmax of (logit - log q) over kept tokens ----------
  // q is streamed non-temporally so the L2-resident logits aren't evicted.
  const float tp = s_tp;
  float best = -INFINITY;
  int   bidx = 0x7fffffff;
  for (int i = tid; i < V4; i += NT) {
    f4 x  = L4[i];
    f4 qv = __builtin_nontemporal_load(Q4 + i);   // TH=NT b128 load
    const int base = i * 4;
    if (x.x >= tp) { float s = x.x - logf(qv.x); if (s > best) { best = s; bidx = base;     } }
    if (x.y >= tp) { float s = x.y - logf(qv.y); if (s > best) { best = s; bidx = base + 1; } }
    if (x.z >= tp) { float s = x.z - logf(qv.z); if (s > best) { best = s; bidx = base + 2; } }
    if (x.w >= tp) { float s = x.w - logf(qv.w); if (s > best) { best = s; bidx = base + 3; } }
  }
  // wave32 reduce, first-index tie-break
  for (int off = 16; off > 0; off >>= 1) {
    float os = __shfl_xor(best, off, 32);
    int   oi = __shfl_xor(bidx, off, 32);
    if (os > best || (os == best && oi < bidx)) { best = os; bidx = oi; }
  }
  if (lane == 0) { rf[wv] = best; ri[wv] = bidx; }
  __syncthreads();
  if (tid == 0) {
    float bs = rf[0];
    int   bi = ri[0];
    for (int w = 1; w < NWAVES; ++w)
      if (rf[w] > bs || (rf[w] == bs && ri[w] < bi)) { bs = rf[w]; bi = ri[w]; }
    out[row] = (float)bi;
  }
}

extern "C" void kernel_launch(void* const* d_in, const int* in_sizes, int n_in,
                              void* d_out, int out_size, void* d_ws, size_t ws_size,
                              hipStream_t stream) {
  (void)in_sizes; (void)n_in; (void)d_ws; (void)ws_size; (void)out_size;
  const float* logits = (const float*)d_in[0];
  const int*   k      = (const int*)d_in[1];
  const float* p      = (const float*)d_in[2];
  const float* q      = (const float*)d_in[3];
  float*       out    = (float*)d_out;
  topk_topp_sample_kernel<<<BATCH, NT, 0, stream>>>(logits, k, p, q, out);
}